// PathfinderDiscoveryNetwork_14748917694810
// MI455X (gfx1250) — compile-verified
//
#include <hip/hip_runtime.h>
#include <math.h>

typedef __attribute__((ext_vector_type(2))) float v2f;
typedef __attribute__((ext_vector_type(8))) float v8f;

#define N_NODES 50000
#define N_EDGES 800000
#define DF 128
#define EF 16
#define EH 64
#define NH 128
#define NC 40
#define NCP 48   // NC padded to 3 WMMA tiles of 16

// ---------------------------------------------------------------- utilities
__global__ void zero_kernel(float* __restrict__ p, int n) {
    int i = blockIdx.x * blockDim.x + threadIdx.x;
    if (i < n) p[i] = 0.0f;
}

// ---------------------------------------------------------------- edge MLP
// ew[e] = sigmoid( relu(edge_feat[e] @ W1 + b1) @ W2 + b2 )
__global__ void __launch_bounds__(256)
edge_mlp_kernel(const float* __restrict__ ef,
                const float* __restrict__ W1, const float* __restrict__ b1,
                const float* __restrict__ W2, const float* __restrict__ b2,
                float* __restrict__ ew, int E) {
    __shared__ float sW1[EF * EH];
    __shared__ float sb1[EH];
    __shared__ float sW2[EH];
    __shared__ float sb2;
    for (int i = threadIdx.x; i < EF * EH; i += blockDim.x) sW1[i] = W1[i];
    if (threadIdx.x < EH) { sb1[threadIdx.x] = b1[threadIdx.x]; sW2[threadIdx.x] = W2[threadIdx.x]; }
    if (threadIdx.x == 0) sb2 = b2[0];
    __syncthreads();

    int e = blockIdx.x * blockDim.x + threadIdx.x;
    if (e >= E) return;

    float f[EF];
    const float4* fr = (const float4*)(ef + (size_t)e * EF);
#pragma unroll
    for (int i = 0; i < 4; ++i) {
        float4 v = fr[i];
        f[4*i+0] = v.x; f[4*i+1] = v.y; f[4*i+2] = v.z; f[4*i+3] = v.w;
    }
    float out = sb2;
#pragma unroll 4
    for (int j = 0; j < EH; ++j) {
        float hj = sb1[j];
#pragma unroll
        for (int i = 0; i < EF; ++i) hj = fmaf(f[i], sW1[i * EH + j], hj);
        hj = fmaxf(hj, 0.0f);
        out = fmaf(hj, sW2[j], out);
    }
    ew[e] = 1.0f / (1.0f + expf(-out));
}

// ---------------------------------------------------------------- GEMM1 (WMMA fp32)
// h[N,128] = x[N,128] @ Wc1[128,128].  8 waves/block, 16 rows/wave, Wc1 in LDS.
__global__ void __launch_bounds__(256)
gemm1_kernel(const float* __restrict__ x, const float* __restrict__ Wc1,
             float* __restrict__ h, int n) {
    __shared__ float Bs[DF * NH];                       // 64 KB
    for (int i = threadIdx.x; i < DF * NH; i += 256) Bs[i] = Wc1[i];
    __syncthreads();

    const int wave = threadIdx.x >> 5;
    const int lane = threadIdx.x & 31;
    const int row_base = blockIdx.x * 128 + wave * 16;
    if (row_base >= n) return;

    const int m    = lane & 15;                         // M (A) / N (B,C) index
    const int kq   = (lane >> 4) << 1;                  // K pair: 0 or 2
    const int half = lane >> 4;

    int row = row_base + m; if (row >= n) row = n - 1;  // clamp (tail safety)
    const float* xr = x + (size_t)row * DF;

    v8f acc[8] = {};
    for (int k = 0; k < DF; k += 4) {
        v2f a; a.x = xr[k + kq]; a.y = xr[k + kq + 1];
        const float* b0 = &Bs[(k + kq)     * NH];
        const float* b1 = &Bs[(k + kq + 1) * NH];
#pragma unroll
        for (int t = 0; t < 8; ++t) {
            int col = t * 16 + m;
            v2f b; b.x = b0[col]; b.y = b1[col];
            acc[t] = __builtin_amdgcn_wmma_f32_16x16x4_f32(
                false, a, false, b, (short)0, acc[t], false, false);
        }
    }

    // Epilogue: fast path when the whole 16-row wave tile is in range
    float* hb = h + (size_t)(row_base + half * 8) * NH + m;
    if (row_base + 16 <= n) {
#pragma unroll
        for (int t = 0; t < 8; ++t) {
#pragma unroll
            for (int r = 0; r < 8; ++r)
                hb[(size_t)r * NH + t * 16] = acc[t][r];
        }
    } else {
#pragma unroll
        for (int t = 0; t < 8; ++t) {
#pragma unroll
            for (int r = 0; r < 8; ++r) {
                int orow = row_base + r + half * 8;
                if (orow < n) h[(size_t)orow * NH + t * 16 + m] = acc[t][r];
            }
        }
    }
}

// ---------------------------------------------------------------- scatter 1
// agg[dst[e], f] += h[src[e], f] * ew[e]   (thread = (edge, feature))
__global__ void __launch_bounds__(256)
scatter1_kernel(const int* __restrict__ ei, const float* __restrict__ ew,
                const float* __restrict__ h, float* __restrict__ agg, int E) {
    int idx = blockIdx.x * blockDim.x + threadIdx.x;
    int e = idx >> 7, f = idx & 127;
    if (e >= E) return;
    float w = ew[e];
    int d = ei[e];            // row 0 = dst
    int s = ei[E + e];        // row 1 = src
    atomicAdd(&agg[(size_t)d * NH + f], h[(size_t)s * NH + f] * w);
}

// ---------------------------------------------------------------- bias + relu (in place)
__global__ void bias_relu_kernel(float* __restrict__ agg, const float* __restrict__ bc1, int n) {
    int i = blockIdx.x * blockDim.x + threadIdx.x;
    if (i < n) agg[i] = fmaxf(agg[i] + bc1[i & 127], 0.0f);
}

// ---------------------------------------------------------------- GEMM2 (WMMA fp32)
// h2[N,40] = h1[N,128] @ Wc2[128,40], N-dim padded to 48 in LDS.
__global__ void __launch_bounds__(256)
gemm2_kernel(const float* __restrict__ h1, const float* __restrict__ Wc2,
             float* __restrict__ h2, int n) {
    __shared__ float Bs[DF * NCP];                      // 24 KB, zero-padded cols 40..47
    for (int i = threadIdx.x; i < DF * NCP; i += 256) {
        int r = i / NCP, c = i % NCP;
        Bs[i] = (c < NC) ? Wc2[r * NC + c] : 0.0f;
    }
    __syncthreads();

    const int wave = threadIdx.x >> 5;
    const int lane = threadIdx.x & 31;
    const int row_base = blockIdx.x * 128 + wave * 16;
    if (row_base >= n) return;

    const int m    = lane & 15;
    const int kq   = (lane >> 4) << 1;
    const int half = lane >> 4;

    int row = row_base + m; if (row >= n) row = n - 1;
    const float* xr = h1 + (size_t)row * DF;

    v8f acc[3] = {};
    for (int k = 0; k < DF; k += 4) {
        v2f a; a.x = xr[k + kq]; a.y = xr[k + kq + 1];
        const float* b0 = &Bs[(k + kq)     * NCP];
        const float* b1 = &Bs[(k + kq + 1) * NCP];
#pragma unroll
        for (int t = 0; t < 3; ++t) {
            int col = t * 16 + m;
            v2f b; b.x = b0[col]; b.y = b1[col];
            acc[t] = __builtin_amdgcn_wmma_f32_16x16x4_f32(
                false, a, false, b, (short)0, acc[t], false, false);
        }
    }

    // tiles 0,1 (cols 0..31) are always fully inside NC=40; tile 2 masks cols.
    const bool colok = (32 + m) < NC;                   // tile-2 column validity
    float* ob = h2 + (size_t)(row_base + half * 8) * NC + m;
    if (row_base + 16 <= n) {
#pragma unroll
        for (int t = 0; t < 2; ++t) {
#pragma unroll
            for (int r = 0; r < 8; ++r)
                ob[(size_t)r * NC + t * 16] = acc[t][r];
        }
        if (colok) {
#pragma unroll
            for (int r = 0; r < 8; ++r)
                ob[(size_t)r * NC + 32] = acc[2][r];
        }
    } else {
#pragma unroll
        for (int t = 0; t < 3; ++t) {
#pragma unroll
            for (int r = 0; r < 8; ++r) {
                int orow = row_base + r + half * 8;
                int ocol = t * 16 + m;
                if (orow < n && ocol < NC) h2[(size_t)orow * NC + ocol] = acc[t][r];
            }
        }
    }
}

// ---------------------------------------------------------------- scatter 2
// out[dst[e], c] += h2[src[e], c] * ew[e]   (class dim padded to 64 lanes)
__global__ void __launch_bounds__(256)
scatter2_kernel(const int* __restrict__ ei, const float* __restrict__ ew,
                const float* __restrict__ h2, float* __restrict__ out, int E) {
    int idx = blockIdx.x * blockDim.x + threadIdx.x;
    int e = idx >> 6, c = idx & 63;
    if (e >= E || c >= NC) return;
    float w = ew[e];
    int d = ei[e];
    int s = ei[E + e];
    atomicAdd(&out[(size_t)d * NC + c], h2[(size_t)s * NC + c] * w);
}

// ---------------------------------------------------------------- -log_softmax (row-wise, in place, +bc2)
__global__ void logsoftmax_kernel(float* __restrict__ out, const float* __restrict__ bc2, int n) {
    int row = blockIdx.x * blockDim.x + threadIdx.x;
    if (row >= n) return;
    float z[NC];
    float mx = -3.4e38f;
#pragma unroll
    for (int j = 0; j < NC; ++j) {
        z[j] = out[(size_t)row * NC + j] + bc2[j];
        mx = fmaxf(mx, z[j]);
    }
    float s = 0.0f;
#pragma unroll
    for (int j = 0; j < NC; ++j) s += expf(z[j] - mx);
    float lse = mx + logf(s);
#pragma unroll
    for (int j = 0; j < NC; ++j) out[(size_t)row * NC + j] = lse - z[j];
}

// ---------------------------------------------------------------- launch
extern "C" void kernel_launch(void* const* d_in, const int* in_sizes, int n_in,
                              void* d_out, int out_size, void* d_ws, size_t ws_size,
                              hipStream_t stream) {
    const float* x    = (const float*)d_in[0];
    const float* efeat= (const float*)d_in[1];
    const float* W1   = (const float*)d_in[2];
    const float* b1   = (const float*)d_in[3];
    const float* W2   = (const float*)d_in[4];
    const float* b2   = (const float*)d_in[5];
    const float* Wc1  = (const float*)d_in[6];
    const float* bc1  = (const float*)d_in[7];
    const float* Wc2  = (const float*)d_in[8];
    const float* bc2  = (const float*)d_in[9];
    const int*   ei   = (const int*)  d_in[10];
    float*       out  = (float*)d_out;

    const int N = N_NODES, E = N_EDGES;

    // workspace layout (floats): ew[E] | h[N*128] (reused as h2[N*40]) | agg[N*128]
    float* wsf = (float*)d_ws;
    float* ew  = wsf;
    float* h   = ew + E;
    float* agg = h + (size_t)N * NH;
    float* h2  = h;   // reuse: scatter1 (last reader of h) precedes gemm2 in stream order

    // 1. zero accumulators (d_out/d_ws are poisoned by harness)
    zero_kernel<<<(N * NH + 255) / 256, 256, 0, stream>>>(agg, N * NH);
    zero_kernel<<<(N * NC + 255) / 256, 256, 0, stream>>>(out, N * NC);

    // 2. per-edge weights
    edge_mlp_kernel<<<(E + 255) / 256, 256, 0, stream>>>(efeat, W1, b1, W2, b2, ew, E);

    // 3. h = x @ Wc1  (WMMA fp32)
    gemm1_kernel<<<(N + 127) / 128, 256, 0, stream>>>(x, Wc1, h, N);

    // 4. agg[dst] += ew * h[src]
    scatter1_kernel<<<(E * 128) / 256, 256, 0, stream>>>(ei, ew, h, agg, E);

    // 5. h1 = relu(agg + bc1), in place
    bias_relu_kernel<<<(N * NH + 255) / 256, 256, 0, stream>>>(agg, bc1, N * NH);

    // 6. h2 = h1 @ Wc2  (WMMA fp32)
    gemm2_kernel<<<(N + 127) / 128, 256, 0, stream>>>(agg, Wc2, h2, N);

    // 7. out[dst] += ew * h2[src]
    scatter2_kernel<<<(E * 64) / 256, 256, 0, stream>>>(ei, ew, h2, out, E);

    // 8. out = -log_softmax(out + bc2)
    logsoftmax_kernel<<<(N + 255) / 256, 256, 0, stream>>>(out, bc2, N);
}